// SoftGlidingBoxesLayerDepracated_50319836839991
// MI455X (gfx1250) — compile-verified
//
#include <hip/hip_runtime.h>
#include <stdint.h>

// ---- CDNA5 TDM availability (device pass only; host pass parses fallback) ----
#if defined(__HIP_DEVICE_COMPILE__) && defined(__gfx1250__)
#  if __has_builtin(__builtin_amdgcn_tensor_load_to_lds)
#    define HAVE_TDM 1
#  endif
#endif
#ifndef HAVE_TDM
#  define HAVE_TDM 0
#endif

// ---- problem constants (match reference) ----
constexpr int kB  = 16;
constexpr int kH  = 256;
constexpr int kW  = 256;
constexpr int kC  = 3;
constexpr int kL  = 8;
constexpr int kS  = 8;                       // max_scale
constexpr int kCH = kC * kL;                 // 24 fused channels (c*L + l)
constexpr int kRowElems = kW * kCH;          // 6144 dwords per (b,h) row
constexpr int kPadRow   = kRowElems + (kRowElems >> 6); // +1 dword pad / 64 dwords
constexpr int kRows     = 16;                // window-start rows owned per block
constexpr int kThreads  = kCH * 32;          // 768 = 24 waves; wave==channel
constexpr int kOut      = kB * kCH * kS;     // 3072 outputs

__device__ __forceinline__ int pidx(int i) { return i + (i >> 6); }

#if HAVE_TDM
typedef unsigned int v4u __attribute__((ext_vector_type(4)));
typedef int          v8i __attribute__((ext_vector_type(8)));
typedef int          v4i __attribute__((ext_vector_type(4)));

// One D# describing a 1 x 6144 dword tile (one fused row), loaded to LDS with
// TDM padding: 1 dword inserted per 64 dwords so LDS lane stride becomes
// 195 dwords (coprime with 64 banks -> conflict-free strided reads).
__device__ __forceinline__ void tdm_issue_row(const float* gsrc, uint32_t lds_off) {
  uint64_t ga = (uint64_t)(uintptr_t)gsrc;
  v4u g0;
  g0.x = 1u;                                  // count=1, user descriptor
  g0.y = lds_off;                             // LDS byte address (flat low 32b)
  g0.z = (uint32_t)ga;                        // global_addr[31:0]
  g0.w = (uint32_t)(ga >> 32) | (2u << 30);   // global_addr[56:32] | type=2
  v8i g1;
  g1[0] = (int)((2u << 16)                    // data_size = 4 bytes
              | (1u << 20)                    // pad_enable
              | (5u << 22));                  // pad_interval = 64 dwords (amount=1)
  g1[1] = (int)((unsigned)(kRowElems & 0xFFFF) << 16); // tensor_dim0 lo16
  g1[2] = (int)(1u << 16);                    // tensor_dim0 hi16=0 | tensor_dim1 lo16=1
  g1[3] = (int)((unsigned)kRowElems << 16);   // tensor_dim1 hi16=0 | tile_dim0=6144
  g1[4] = 1;                                  // tile_dim1=1, tile_dim2=0
  g1[5] = kRowElems;                          // tensor_dim0_stride lo32
  g1[6] = 0;
  g1[7] = 0;
  v4i z4 = {0, 0, 0, 0};                      // 2D tensor: groups 2/3 unused
  v8i z8 = {0, 0, 0, 0, 0, 0, 0, 0};          // extra group (6-arg toolchain form)
  __builtin_amdgcn_tensor_load_to_lds(g0, g1, z4, z4, z8, /*cpol*/0);
}
#endif

__device__ __forceinline__ void stage_issue(const float* gsrc, float* dst_lds) {
#if HAVE_TDM
  if (threadIdx.x == 0)
    tdm_issue_row(gsrc, (uint32_t)(uintptr_t)dst_lds);
#else
  for (int i = (int)threadIdx.x; i < kRowElems; i += kThreads)
    dst_lds[pidx(i)] = gsrc[i];
#endif
}

__device__ __forceinline__ void stage_commit() {
#if HAVE_TDM
  __builtin_amdgcn_s_wait_tensorcnt(0);       // s_wait_tensorcnt 0
#endif
  __syncthreads();
}

// Block = (b, strip of kRows window-start rows). Wave = channel, lane owns
// 8 consecutive w columns. Vertical window maxes via in-register recurrence
// m_s(r) = max(x(r), m_{s-1}(r-1)); horizontal via 8 in-register columns +
// <=7 shuffled halo columns from lane+1.
__global__ __launch_bounds__(kThreads, 1)
void pool_strip_kernel(const float* __restrict__ x, float* __restrict__ acc_ws) {
  __shared__ float rowbuf[2][kPadRow];

  const int b       = blockIdx.y;
  const int r0      = blockIdx.x * kRows;
  const int lane    = threadIdx.x & 31;
  const int ch      = threadIdx.x >> 5;                 // 0..23
  const int rEnd    = min(r0 + kRows + kS - 1, kH);     // last input row + 1
  const int rOwnEnd = min(r0 + kRows, kH);              // owned window starts

  const float* xb = x + (size_t)b * kH * kRowElems;

  float v[8][8];                                        // v[col][s-1]
  #pragma unroll
  for (int c = 0; c < 8; ++c)
    #pragma unroll
    for (int s = 0; s < 8; ++s) v[c][s] = -__builtin_inff();
  float acc[8] = {0.f, 0.f, 0.f, 0.f, 0.f, 0.f, 0.f, 0.f};

  stage_issue(xb + (size_t)r0 * kRowElems, rowbuf[0]);
  stage_commit();

  for (int r = r0; r < rEnd; ++r) {
    const int slot = (r - r0) & 1;
    if (r + 1 < rEnd)                                   // prefetch next row
      stage_issue(xb + (size_t)(r + 1) * kRowElems, rowbuf[slot ^ 1]);

    // --- vertical running maxes (in place, descending s) ---
    #pragma unroll
    for (int c = 0; c < 8; ++c) {
      const int   w  = lane * 8 + c;
      const float xv = rowbuf[slot][pidx(w * kCH + ch)];
      #pragma unroll
      for (int s = 7; s >= 1; --s) v[c][s] = fmaxf(xv, v[c][s - 1]);
      v[c][0] = xv;
    }

    // --- horizontal s-window + accumulate (windows ending at row r) ---
    #pragma unroll
    for (int s = 1; s <= 8; ++s) {
      const int i0 = r - s + 1;                         // window start row
      if (i0 >= r0 && i0 < rOwnEnd) {                   // uniform branch
        float hal[7];
        #pragma unroll
        for (int j = 0; j + 1 < s; ++j)                 // neighbor cols 0..s-2
          hal[j] = __shfl_down(v[j][s - 1], 1, 32);
        #pragma unroll
        for (int c = 0; c < 8; ++c) {
          float m = v[c][s - 1];
          #pragma unroll
          for (int k = 1; k < s; ++k) {
            const int cc = c + k;
            const float nb = (cc < 8) ? v[cc & 7][s - 1] : hal[cc & 7];
            m = fmaxf(m, nb);
          }
          if (lane * 8 + c + s <= kW)                   // right-edge validity
            acc[s - 1] += m;
        }
      }
    }
    stage_commit();                                     // next row ready
  }

  // --- wave reduction (lanes cover disjoint w) + global accumulate ---
  #pragma unroll
  for (int s = 0; s < 8; ++s) {
    float a = acc[s];
    #pragma unroll
    for (int off = 16; off >= 1; off >>= 1)
      a += __shfl_xor(a, off, 32);
    if (lane == 0)
      atomicAdd(&acc_ws[(b * kCH + ch) * kS + s], a);
  }
}

__global__ void zero_ws_kernel(float* __restrict__ ws) {
  const int i = blockIdx.x * blockDim.x + threadIdx.x;
  if (i < kOut) ws[i] = 0.f;
}

__global__ void finalize_kernel(const float* __restrict__ ws, float* __restrict__ out) {
  const int i = blockIdx.x * blockDim.x + threadIdx.x;
  if (i < kOut) out[i] = fmaxf(ws[i], 0.f) + 1.f;       // relu(membership) + 1
}

extern "C" void kernel_launch(void* const* d_in, const int* in_sizes, int n_in,
                              void* d_out, int out_size, void* d_ws, size_t ws_size,
                              hipStream_t stream) {
  (void)in_sizes; (void)n_in; (void)out_size; (void)ws_size;
  const float* x   = (const float*)d_in[0];   // [B,H,W,C,L] f32
  float*       ws  = (float*)d_ws;            // 3072-float accumulator
  float*       out = (float*)d_out;           // [B,C,L,S] f32

  zero_ws_kernel<<<dim3((kOut + 255) / 256), dim3(256), 0, stream>>>(ws);
  dim3 grid(kH / kRows, kB);                  // 16 x 16 = 256 blocks
  pool_strip_kernel<<<grid, dim3(kThreads), 0, stream>>>(x, ws);
  finalize_kernel<<<dim3((kOut + 255) / 256), dim3(256), 0, stream>>>(ws, out);
}